// SmallSMBlock_4286377361802
// MI455X (gfx1250) — compile-verified
//
#include <hip/hip_runtime.h>
#include <hip/hip_bf16.h>

// ---------------------------------------------------------------------------
// Spatially-varying 3x3 conv, fused:
//   Phase 1: K = Conv2d(image, W_KL) + b  computed with v_wmma_f32_16x16x4_f32
//            (9x9 weights x 9xN im2col patches, padded to 16x12), K kept in LDS.
//   Phase 2: y[b,h,w] = sum_t K[t,h,w] * x[b, h+t/3-1, w+t%3-1]
//            x tiles streamed HBM->LDS with global_load_async_to_lds_b128,
//            double buffered (prefetch batch b+1 while computing batch b).
// Bandwidth-bound: ~135 MB total HBM traffic -> ~5.8 us floor at 23.3 TB/s.
// ---------------------------------------------------------------------------

typedef __attribute__((ext_vector_type(2))) float v2f;
typedef __attribute__((ext_vector_type(4))) float vf4;
typedef __attribute__((ext_vector_type(8))) float v8f;

#define HH     1024
#define WW     1024
#define NB     16
#define TH     16               // tile rows per block
#define TW     64               // tile cols per block
#define XROWS  (TH + 2)         // halo tile rows = 18
#define XPITCH 72               // LDS cols: 4 left halo + 64 + 2 right + 2 pad
                                // (base at w0-4 keeps global B128 loads 16B aligned)
#define NF4    (XROWS * (XPITCH / 4))   // 18 * 18 = 324 float4 per halo tile
#define HALO   3                // pixel (ph,pw) tap (a,d) -> LDS col pw + 3 + d

// Flat 'shared' pointer low 32 bits are the LDS byte offset on gfx1250
__device__ __forceinline__ uint32_t lds_off32(const void* p) {
  return (uint32_t)(uintptr_t)p;
}

// Async HBM -> LDS tile copy, 16B granules (interior tiles only).
// srcBase points at (h0-1, w0-4): 16B aligned because w0 % 64 == 0.
// Tracked by ASYNCcnt; caller pairs with s_wait_asynccnt + barrier.
__device__ __forceinline__ void load_tile_async(float* dstLds, const float* srcBase) {
  const int tid = threadIdx.x;
  const uint32_t dbase = lds_off32(dstLds);
  #pragma unroll
  for (int k = 0; k < 2; ++k) {            // ceil(324/256) = 2
    int idx = tid + k * 256;
    if (idx < NF4) {
      int r  = idx / (XPITCH / 4);
      int c4 = idx - r * (XPITCH / 4);
      uint32_t laddr = dbase + (uint32_t)(r * XPITCH + c4 * 4) * 4u;
      int goff = (r * WW + c4 * 4) * 4;    // byte offset from srcBase, 16B aligned
      asm volatile("global_load_async_to_lds_b128 %0, %1, %2"
                   :: "v"(laddr), "v"(goff), "s"(srcBase) : "memory");
    }
  }
}

// Bounds-checked fallback for image-border tiles (zero padding), plain DS path.
__device__ __forceinline__ void load_tile_edge(float* dstLds, const float* src,
                                               int h0, int w0) {
  const int tid = threadIdx.x;
  for (int idx = tid; idx < XROWS * XPITCH; idx += 256) {
    int r = idx / XPITCH;
    int c = idx - r * XPITCH;
    int gh = h0 - 1 + r;
    int gw = w0 - 4 + c;
    float v = 0.0f;
    if (gh >= 0 && gh < HH && gw >= 0 && gw < WW) v = src[(size_t)gh * WW + gw];
    dstLds[r * XPITCH + c] = v;
  }
}

__global__ __launch_bounds__(256)
void svconv_fused_kernel(const float* __restrict__ image,
                         const float* __restrict__ x,
                         const float* __restrict__ Wk,   // (9,1,3,3) flat
                         const float* __restrict__ bk,   // (9,)
                         float* __restrict__ y) {
  __shared__ float Klds[9 * TH * TW];          // 36 KB: per-pixel kernels
  __shared__ float xs[2][XROWS * XPITCH];      // 2 x 5.1 KB: halo tiles

  const int h0 = blockIdx.y * TH;
  const int w0 = blockIdx.x * TW;
  const bool interior = (h0 > 0) && (h0 + TH < HH) && (w0 >= 4) && (w0 + TW < WW);
  const int tid  = threadIdx.x;
  const int lane = tid & 31;
  const int wv   = tid >> 5;

  // ---- Load image halo tile into xs[0] -----------------------------------
  if (interior) {
    load_tile_async(xs[0], image + (size_t)(h0 - 1) * WW + (w0 - 4));
    asm volatile("s_wait_asynccnt 0x0" ::: "memory");
  } else {
    load_tile_edge(xs[0], image, h0, w0);
  }
  __syncthreads();

  // ---- Kick off x batch 0 into xs[1]; it streams while WMMA computes K ----
  if (interior) {
    load_tile_async(xs[1], x + (size_t)(h0 - 1) * WW + (w0 - 4));
  } else {
    load_tile_edge(xs[1], x, h0, w0);
  }

  // ---- Phase 1: K = conv(image) as GEMM, A(16x12 weights) x B(12x16 patch)
  // A-operand layout (f32 16x16x4): lanes 0-15 hold K={k0,k0+1}, lanes 16-31
  // hold K={k0+2,k0+3}; rows M = output channel (9 valid, rest zero-padded).
  const int mrow  = lane & 15;
  const int khalf = (lane >> 4) << 1;        // 0 or 2
  v2f A0, A1, A2;
  {
    const bool mv = (mrow < 9);
    int t = 0 + khalf;
    A0.x = mv ? Wk[mrow * 9 + t] : 0.0f;
    A0.y = mv ? Wk[mrow * 9 + t + 1] : 0.0f;
    t = 4 + khalf;
    A1.x = mv ? Wk[mrow * 9 + t] : 0.0f;
    A1.y = mv ? Wk[mrow * 9 + t + 1] : 0.0f;
    t = 8 + khalf;                            // taps 9..11 are zero padding
    A2.x = (mv && t < 9) ? Wk[mrow * 9 + t] : 0.0f;
    A2.y = 0.0f;
  }
  // Hoist uniform bias reads out of the group loop (stay in SGPRs).
  float bias[9];
  #pragma unroll
  for (int r = 0; r < 9; ++r) bias[r] = bk[r];

  const float* img = xs[0];
  #pragma unroll
  for (int g = 0; g < 8; ++g) {              // 8 waves x 8 groups x 16 px = 1024
    const int p  = (wv * 8 + g) * 16 + (lane & 15);
    const int ph = p >> 6;
    const int pw = p & 63;
    // B-operand: column N = pixel (lane&15), row K = tap t; lanes 0-15 carry
    // rows {k0,k0+1}, lanes 16-31 rows {k0+2,k0+3} (mirrors A striping).
    v2f B0, B1, B2;
    int t;
    t = 0 + khalf; B0.x = img[(ph + t / 3) * XPITCH + pw + HALO + t % 3];
    t = 1 + khalf; B0.y = img[(ph + t / 3) * XPITCH + pw + HALO + t % 3];
    t = 4 + khalf; B1.x = img[(ph + t / 3) * XPITCH + pw + HALO + t % 3];
    t = 5 + khalf; B1.y = img[(ph + t / 3) * XPITCH + pw + HALO + t % 3];
    t = 8 + khalf;
    B2.x = (t < 9) ? img[(ph + t / 3) * XPITCH + pw + HALO + t % 3] : 0.0f;
    B2.y = 0.0f;

    v8f acc = {};
    acc = __builtin_amdgcn_wmma_f32_16x16x4_f32(false, A0, false, B0,
                                                (short)0, acc, false, false);
    acc = __builtin_amdgcn_wmma_f32_16x16x4_f32(false, A1, false, B1,
                                                (short)0, acc, false, false);
    acc = __builtin_amdgcn_wmma_f32_16x16x4_f32(false, A2, false, B2,
                                                (short)0, acc, false, false);

    // C/D layout: VGPR r, lanes 0-15 -> M=r ; lanes 16-31 -> M=r+8.
    if (lane < 16) {
      #pragma unroll
      for (int r = 0; r < 8; ++r)
        Klds[(r * TH + ph) * TW + pw] = acc[r] + bias[r];
    } else {
      Klds[(8 * TH + ph) * TW + pw] = acc[0] + bias[8];
    }
  }

  asm volatile("s_wait_asynccnt 0x0" ::: "memory");   // x batch 0 tile landed
  __syncthreads();                                    // K + tile visible to all

  // ---- Phase 2: stream 16 batches, double-buffered async tiles ------------
  const int ph2 = tid >> 4;            // 0..15
  const int pw2 = (tid & 15) << 2;     // 0..60, float4 of pixels per thread
  #pragma unroll 1
  for (int b = 0; b < NB; ++b) {
    const float* cur = xs[(b + 1) & 1];
    if (b + 1 < NB) {                  // prefetch next batch into the free buf
      float* nbuf = xs[b & 1];
      const float* src = x + (size_t)(b + 1) * HH * WW;
      if (interior) load_tile_async(nbuf, src + (size_t)(h0 - 1) * WW + (w0 - 4));
      else          load_tile_edge(nbuf, src, h0, w0);
    }

    vf4 acc = {0.0f, 0.0f, 0.0f, 0.0f};
    #pragma unroll
    for (int a = 0; a < 3; ++a) {
      // need LDS cols [pw2+3 .. pw2+8]: one aligned b128 at +4, scalars at +3,+8
      const float* row = cur + (ph2 + a) * XPITCH + pw2;
      vf4 xv = *(const vf4*)(row + 4);                 // cols +4..+7 (16B aligned)
      float rv[6] = {row[3], xv.x, xv.y, xv.z, xv.w, row[8]};
      #pragma unroll
      for (int d = 0; d < 3; ++d) {
        vf4 kv = *(const vf4*)&Klds[((3 * a + d) * TH + ph2) * TW + pw2];
        acc.x += kv.x * rv[d + 0];
        acc.y += kv.y * rv[d + 1];
        acc.z += kv.z * rv[d + 2];
        acc.w += kv.w * rv[d + 3];
      }
    }

    float* yp = y + (size_t)b * HH * WW + (size_t)(h0 + ph2) * WW + (w0 + pw2);
    *(vf4*)yp = acc;                    // coalesced 16B store

    asm volatile("s_wait_asynccnt 0x0" ::: "memory");  // next tile complete
    __syncthreads();                                   // all waves done reading
  }
}

extern "C" void kernel_launch(void* const* d_in, const int* in_sizes, int n_in,
                              void* d_out, int out_size, void* d_ws, size_t ws_size,
                              hipStream_t stream) {
  const float* image = (const float*)d_in[0];  // (1,1024,1024)
  const float* x     = (const float*)d_in[1];  // (16,1,1024,1024)
  const float* Wk    = (const float*)d_in[2];  // (9,1,3,3)
  const float* bk    = (const float*)d_in[3];  // (9,)
  float*       y     = (float*)d_out;          // (16,1,1024,1024)

  dim3 grid(WW / TW, HH / TH);                 // (16, 64) blocks
  svconv_fused_kernel<<<grid, 256, 0, stream>>>(image, x, Wk, bk, y);
}